// LIF_29326036697290
// MI455X (gfx1250) — compile-verified
//
#include <hip/hip_runtime.h>

// LIF temporal scan, bandwidth-bound streaming kernel for gfx1250 (MI455X).
// x: (T, B, D) f32, u0: (B, D) f32, tau_mem: (D,) f32
// out: spike (T,B,D) f32 followed by mem (T,B,D) f32.
//
// Roofline: ~403 MB moved (134 MB x-in, 268 MB out), ~0.23 GFLOP total
// -> pure HBM-bound (~17 us floor @ 23.3 TB/s). No matrix ops exist in the
// recurrence, so no WMMA; the win is coalesced B128 NT traffic + prefetch.

typedef __attribute__((ext_vector_type(4))) float v4f;

#define VTHR 1.0f

__global__ __launch_bounds__(256) void lif_scan_kernel(
    const v4f* __restrict__ x,        // T * N4 float4s
    const v4f* __restrict__ u0,       // N4 float4s
    const v4f* __restrict__ tau,      // D4 float4s
    v4f* __restrict__ spike_out,      // T * N4 float4s
    v4f* __restrict__ mem_out,        // T * N4 float4s
    int T, int N4, int D4, int d4mask /* D4-1 if pow2, else -1 */)
{
    const int idx = blockIdx.x * blockDim.x + threadIdx.x;
    if (idx >= N4) return;

    // Per-channel decay, computed once per thread (v_exp_f32, stays in VGPRs).
    const int ch = (d4mask >= 0) ? (idx & d4mask) : (idx % D4);
    const v4f tv = tau[ch];
    v4f a;
    a.x = __expf(-1.0f / fabsf(tv.x));
    a.y = __expf(-1.0f / fabsf(tv.y));
    a.z = __expf(-1.0f / fabsf(tv.z));
    a.w = __expf(-1.0f / fabsf(tv.w));
    const v4f oma = 1.0f - a;

    v4f u = u0[idx];
    v4f z = {0.0f, 0.0f, 0.0f, 0.0f};

    const v4f* xp = x + idx;
    v4f* sp = spike_out + idx;
    v4f* mp = mem_out + idx;

    // Depth-2 software pipeline: two B128 loads in flight per wave, plus
    // counter-free global_prefetch_b8 running 5 timesteps ahead.
    v4f xv = __builtin_nontemporal_load(xp);
    v4f xn = (T > 1) ? __builtin_nontemporal_load(xp + N4) : xv;

    for (int t = 0; t < T; ++t) {
        if (t + 5 < T)
            __builtin_prefetch((const void*)(xp + (long)(t + 5) * N4), 0, 0);

        v4f x2 = xn;
        if (t + 2 < T)
            x2 = __builtin_nontemporal_load(xp + (long)(t + 2) * N4);

        // u = u*alpha + x*(1-alpha) - z*VTHR
        u.x = fmaf(u.x, a.x, xv.x * oma.x) - z.x * VTHR;
        u.y = fmaf(u.y, a.y, xv.y * oma.y) - z.y * VTHR;
        u.z = fmaf(u.z, a.z, xv.z * oma.z) - z.z * VTHR;
        u.w = fmaf(u.w, a.w, xv.w * oma.w) - z.w * VTHR;

        // z = (u - VTHR > 0) ? 1 : 0
        z.x = (u.x - VTHR > 0.0f) ? 1.0f : 0.0f;
        z.y = (u.y - VTHR > 0.0f) ? 1.0f : 0.0f;
        z.z = (u.z - VTHR > 0.0f) ? 1.0f : 0.0f;
        z.w = (u.w - VTHR > 0.0f) ? 1.0f : 0.0f;

        // mem = max(u / VTHR, 0)
        v4f m;
        m.x = fmaxf(u.x * (1.0f / VTHR), 0.0f);
        m.y = fmaxf(u.y * (1.0f / VTHR), 0.0f);
        m.z = fmaxf(u.z * (1.0f / VTHR), 0.0f);
        m.w = fmaxf(u.w * (1.0f / VTHR), 0.0f);

        const long off = (long)t * N4;
        __builtin_nontemporal_store(z, sp + off);
        __builtin_nontemporal_store(m, mp + off);

        xv = xn;
        xn = x2;
    }
}

extern "C" void kernel_launch(void* const* d_in, const int* in_sizes, int n_in,
                              void* d_out, int out_size, void* d_ws, size_t ws_size,
                              hipStream_t stream) {
    (void)n_in; (void)d_ws; (void)ws_size; (void)out_size;

    const float* x   = (const float*)d_in[0];   // T*B*D
    const float* u0  = (const float*)d_in[1];   // B*D
    const float* tau = (const float*)d_in[2];   // D

    const int TBD = in_sizes[0];
    const int BD  = in_sizes[1];
    const int D   = in_sizes[2];
    const int T   = TBD / BD;
    const int N4  = BD / 4;
    const int D4  = D / 4;
    const int d4mask = ((D4 & (D4 - 1)) == 0) ? (D4 - 1) : -1;

    float* spike = (float*)d_out;               // first T*B*D
    float* mem   = spike + (size_t)TBD;         // second T*B*D

    const int block = 256;
    const int grid  = (N4 + block - 1) / block;

    lif_scan_kernel<<<grid, block, 0, stream>>>(
        (const v4f*)x, (const v4f*)u0, (const v4f*)tau,
        (v4f*)spike, (v4f*)mem, T, N4, D4, d4mask);
}